// DINOLoss_20237885899179
// MI455X (gfx1250) — compile-verified
//
#include <hip/hip_runtime.h>
#include <hip/hip_bf16.h>

#define D        65536
#define B_SZ     64
#define G_CROPS  2
#define T_CROPS  10
#define INV_ST   10.0f      // 1/0.1
#define INV_TT   25.0f      // 1/0.04
#define CHUNK    1024
#define SPLIT    16         // D pieces per batch element -> 64*16 = 1024 blocks
#define PIECE    (D / SPLIT)

typedef float v2f __attribute__((ext_vector_type(2)));
typedef float v8f __attribute__((ext_vector_type(8)));

// ---------------------------------------------------------------- kernel A
// Per-teacher-row: max and 1/sum(exp((x - max) * 25))
__global__ void k_teacher_stats(const float* __restrict__ teacher,
                                float* __restrict__ m_out,
                                float* __restrict__ invS_out) {
    __shared__ float red[256];
    const int tid = threadIdx.x;
    const float* row = teacher + (size_t)blockIdx.x * D;

    float mx = -INFINITY;
    for (int i = tid; i < D; i += 256) mx = fmaxf(mx, row[i]);
    red[tid] = mx; __syncthreads();
    for (int s = 128; s > 0; s >>= 1) {
        if (tid < s) red[tid] = fmaxf(red[tid], red[tid + s]);
        __syncthreads();
    }
    mx = red[0]; __syncthreads();

    float sum = 0.0f;
    for (int i = tid; i < D; i += 256) sum += expf((row[i] - mx) * INV_TT);
    red[tid] = sum; __syncthreads();
    for (int s = 128; s > 0; s >>= 1) {
        if (tid < s) red[tid] += red[tid + s];
        __syncthreads();
    }
    if (tid == 0) { m_out[blockIdx.x] = mx; invS_out[blockIdx.x] = 1.0f / red[0]; }
}

// ---------------------------------------------------------------- kernel B
// new_center[d] = 0.9*center[d] + 0.1 * mean_r softmax(teacher)[r,d]
__global__ void k_center(const float* __restrict__ teacher,
                         const float* __restrict__ center,
                         const float* __restrict__ m,
                         const float* __restrict__ invS,
                         float* __restrict__ newC,
                         float* __restrict__ csum_part) {
    __shared__ float sm[B_SZ * G_CROPS];
    __shared__ float si[B_SZ * G_CROPS];
    __shared__ float red[256];
    const int tid = threadIdx.x;
    if (tid < B_SZ * G_CROPS) { sm[tid] = m[tid]; si[tid] = invS[tid]; }
    __syncthreads();

    const int d = blockIdx.x * 256 + tid;
    float acc = 0.0f;
    for (int r = 0; r < B_SZ * G_CROPS; ++r)
        acc += expf((teacher[(size_t)r * D + d] - sm[r]) * INV_TT) * si[r];
    const float nc = center[d] * 0.9f + acc * (0.1f / (float)(B_SZ * G_CROPS));
    newC[d] = nc;

    red[tid] = nc; __syncthreads();
    for (int s = 128; s > 0; s >>= 1) {
        if (tid < s) red[tid] += red[tid + s];
        __syncthreads();
    }
    if (tid == 0) csum_part[blockIdx.x] = red[0];
}

// ---------------------------------------------------------------- kernel D
// grid (B_SZ, SPLIT). Block (b,p) handles K-range [p*PIECE, (p+1)*PIECE).
//   WMMA tile: C[16,16] += A(16 x K) * B(K x 16)
//     A rows 0,1 = teacher softmax probs, row 2 = new_center, row 3.. = zeros
//     B cols 0..9 = raw student rows (x10 temp scale applied at the end)
//   Fused: online logsumexp of student rows (10 rows x 16 slices of 64 elems).
__global__ void k_dots_wmma(const float* __restrict__ student,
                            const float* __restrict__ teacher,
                            const float* __restrict__ newC,
                            const float* __restrict__ m,
                            const float* __restrict__ invS,
                            float* __restrict__ dotParts,   // [B][SPLIT][48]
                            float* __restrict__ lseM,       // [640][SPLIT]
                            float* __restrict__ lseS) {     // [640][SPLIT]
    __shared__ float lA[4 * CHUNK];          // probs g=0,1 + center + ZERO row
    __shared__ float lB[(T_CROPS + 1) * CHUNK]; // student crops + ZERO row
    __shared__ float lOut[8 * 48];           // 8 waves x (3 rows x 16 cols)
    __shared__ float cm[160];                // per-slice lse max
    __shared__ float cs[160];                // per-slice lse sumexp

    const int b    = blockIdx.x;
    const int p    = blockIdx.y;
    const int tid  = threadIdx.x;
    const int lane = tid & 31;
    const int wave = tid >> 5;
    const int M    = lane & 15;              // A-row index (== B-col index N)
    const int khalf = (lane < 16) ? 0 : 2;   // K split per ISA A/B layout
    const int aRow  = (M < 3) ? M : 3;       // row 3 is a physical zero row
    const int bRow  = (M < T_CROPS) ? M : T_CROPS; // row 10 is zero row

    const float m0 = m[b * G_CROPS + 0], i0 = invS[b * G_CROPS + 0];
    const float m1 = m[b * G_CROPS + 1], i1 = invS[b * G_CROPS + 1];
    const float* t0   = teacher + (size_t)(b * G_CROPS) * D;
    const float* t1   = t0 + D;
    const float* srow = student + (size_t)(b * T_CROPS) * D;

    const int li = tid * 4;                  // each thread: one float4 per row
    // physical zero pad rows (written once; never touched again)
    *(float4*)(&lA[3 * CHUNK + li])       = make_float4(0.f, 0.f, 0.f, 0.f);
    *(float4*)(&lB[T_CROPS * CHUNK + li]) = make_float4(0.f, 0.f, 0.f, 0.f);

    v8f acc[4] = {};

    // fused student LSE state (threads 0..159: row t = tid>>4, slice j = tid&15)
    const int lt = tid >> 4, lj = tid & 15;
    float lm = -INFINITY, ls = 0.0f;

    const int pbase = p * PIECE;
    for (int c0 = pbase; c0 < pbase + PIECE; c0 += CHUNK) {
        float4 v;
        v = *(const float4*)(t0 + c0 + li);
        lA[0 * CHUNK + li + 0] = expf((v.x - m0) * INV_TT) * i0;
        lA[0 * CHUNK + li + 1] = expf((v.y - m0) * INV_TT) * i0;
        lA[0 * CHUNK + li + 2] = expf((v.z - m0) * INV_TT) * i0;
        lA[0 * CHUNK + li + 3] = expf((v.w - m0) * INV_TT) * i0;
        v = *(const float4*)(t1 + c0 + li);
        lA[1 * CHUNK + li + 0] = expf((v.x - m1) * INV_TT) * i1;
        lA[1 * CHUNK + li + 1] = expf((v.y - m1) * INV_TT) * i1;
        lA[1 * CHUNK + li + 2] = expf((v.z - m1) * INV_TT) * i1;
        lA[1 * CHUNK + li + 3] = expf((v.w - m1) * INV_TT) * i1;
        *(float4*)(&lA[2 * CHUNK + li]) = *(const float4*)(newC + c0 + li);
        #pragma unroll
        for (int t = 0; t < T_CROPS; ++t)
            *(float4*)(&lB[t * CHUNK + li]) =
                *(const float4*)(srow + (size_t)t * D + c0 + li);
        __syncthreads();

        // --- WMMA: wave owns a 128-wide K slice; 4 rotating accumulators ---
        const int kb = wave * 128;
        #pragma unroll
        for (int it = 0; it < 32; ++it) {
            const int k = kb + it * 4 + khalf;
            v2f a, bf;
            a.x  = lA[aRow * CHUNK + k];
            a.y  = lA[aRow * CHUNK + k + 1];
            bf.x = lB[bRow * CHUNK + k];
            bf.y = lB[bRow * CHUNK + k + 1];
            acc[it & 3] = __builtin_amdgcn_wmma_f32_16x16x4_f32(
                false, a, false, bf, (short)0, acc[it & 3], false, false);
        }

        // --- fused online LSE over this chunk (rows in LDS) ---
        if (tid < 160) {
            const float* seg = &lB[lt * CHUNK + lj * 64];
            float cmax = lm;
            #pragma unroll 8
            for (int e = 0; e < 64; ++e) cmax = fmaxf(cmax, seg[e]);
            ls *= expf((lm - cmax) * INV_ST);   // rescale old partial
            float add = 0.0f;
            #pragma unroll 8
            for (int e = 0; e < 64; ++e) add += expf((seg[e] - cmax) * INV_ST);
            ls += add;
            lm = cmax;
        }
        __syncthreads();
    }

    // combine 4 accumulators; C/D layout: VGPR j = (M=j, N=lane) for lanes 0-15
    v8f tot = acc[0] + acc[1] + acc[2] + acc[3];
    if (lane < 16) {
        #pragma unroll
        for (int j = 0; j < 3; ++j)
            lOut[wave * 48 + j * 16 + lane] = tot[j];
    }
    if (tid < 160) { cm[tid] = lm; cs[tid] = ls; }
    __syncthreads();

    if (tid < 48) {
        float s = 0.0f;
        #pragma unroll
        for (int w = 0; w < 8; ++w) s += lOut[w * 48 + tid];
        dotParts[(b * SPLIT + p) * 48 + tid] = s;
    }
    if (tid < T_CROPS) {                      // merge 16 slices of row tid
        float M2 = -INFINITY;
        for (int j = 0; j < 16; ++j) M2 = fmaxf(M2, cm[tid * 16 + j]);
        float S2 = 0.0f;
        for (int j = 0; j < 16; ++j)
            S2 += cs[tid * 16 + j] * expf((cm[tid * 16 + j] - M2) * INV_ST);
        lseM[(b * T_CROPS + tid) * SPLIT + p] = M2;
        lseS[(b * T_CROPS + tid) * SPLIT + p] = S2;
    }
}

// ---------------------------------------------------------------- kernel E1a
// Merge LSE pieces -> L[row] (one thread per student row, fixed order)
__global__ void k_lse_merge(const float* __restrict__ lseM,
                            const float* __restrict__ lseS,
                            float* __restrict__ L_out) {
    const int r = blockIdx.x * blockDim.x + threadIdx.x;
    if (r >= B_SZ * T_CROPS) return;
    float M2 = -INFINITY;
    for (int p = 0; p < SPLIT; ++p) M2 = fmaxf(M2, lseM[r * SPLIT + p]);
    float S2 = 0.0f;
    for (int p = 0; p < SPLIT; ++p)
        S2 += lseS[r * SPLIT + p] * expf((lseM[r * SPLIT + p] - M2) * INV_ST);
    L_out[r] = M2 * INV_ST + logf(S2);
}

// ---------------------------------------------------------------- kernel E1b
// Sum dot pieces (fixed order)
__global__ void k_dot_merge(const float* __restrict__ dotParts,
                            float* __restrict__ dotRed) {
    const int b = blockIdx.x, i = threadIdx.x;   // 64 blocks x 48 threads
    float s = 0.0f;
    for (int p = 0; p < SPLIT; ++p) s += dotParts[(b * SPLIT + p) * 48 + i];
    dotRed[b * 48 + i] = s;
}

// ---------------------------------------------------------------- kernel E2
__global__ void k_finalize(const float* __restrict__ dotRed,
                           const float* __restrict__ L,
                           const float* __restrict__ csum_part,
                           float* __restrict__ out) {
    if (blockIdx.x != 0 || threadIdx.x != 0) return;
    float csum = 0.0f;
    for (int i = 0; i < 256; ++i) csum += csum_part[i];
    const float oneMinusC = 1.0f - csum;

    double total = 0.0;
    for (int b = 0; b < B_SZ; ++b) {
        const float* dt = dotRed + b * 48;
        for (int t = 0; t < T_CROPS; ++t) {
            const float w = (t < G_CROPS) ? (float)(G_CROPS - 1) : (float)G_CROPS;
            const float dotC = dt[2 * 16 + t] * INV_ST;       // dot(new_center, s_t)
            total += (double)(w * (dotC + L[b * T_CROPS + t] * oneMinusC));
        }
        for (int g = 0; g < G_CROPS; ++g)
            for (int t = 0; t < T_CROPS; ++t)
                if (t != g)
                    total -= (double)(dt[g * 16 + t] * INV_ST); // dot(P_g, s_t)
    }
    const double n_terms = (double)(B_SZ * G_CROPS * (T_CROPS - 1)); // 1152
    out[0] = (float)(total / n_terms);
}

// ----------------------------------------------------------------
extern "C" void kernel_launch(void* const* d_in, const int* in_sizes, int n_in,
                              void* d_out, int out_size, void* d_ws, size_t ws_size,
                              hipStream_t stream) {
    (void)in_sizes; (void)n_in; (void)out_size; (void)ws_size;
    const float* student = (const float*)d_in[0];   // (640, 65536)
    const float* teacher = (const float*)d_in[1];   // (128, 65536)
    const float* center  = (const float*)d_in[2];   // (1, 65536)
    float* out = (float*)d_out;

    float* ws        = (float*)d_ws;
    float* m         = ws;                          // 128
    float* invS      = m + 128;                     // 128
    float* newC      = invS + 128;                  // 65536
    float* csum_part = newC + D;                    // 256
    float* Lrow      = csum_part + 256;             // 640
    float* lseM      = Lrow + 640;                  // 640*16 = 10240
    float* lseS      = lseM + B_SZ * T_CROPS * SPLIT;   // 10240
    float* dotParts  = lseS + B_SZ * T_CROPS * SPLIT;   // 64*16*48 = 49152
    float* dotRed    = dotParts + B_SZ * SPLIT * 48;    // 3072

    k_teacher_stats<<<B_SZ * G_CROPS, 256, 0, stream>>>(teacher, m, invS);
    k_center<<<D / 256, 256, 0, stream>>>(teacher, center, m, invS, newC, csum_part);
    k_dots_wmma<<<dim3(B_SZ, SPLIT), 256, 0, stream>>>(
        student, teacher, newC, m, invS, dotParts, lseM, lseS);
    k_lse_merge<<<(B_SZ * T_CROPS + 127) / 128, 128, 0, stream>>>(lseM, lseS, Lrow);
    k_dot_merge<<<B_SZ, 48, 0, stream>>>(dotParts, dotRed);
    k_finalize<<<1, 32, 0, stream>>>(dotRed, Lrow, csum_part, out);
}